// TypedBinaryTreeLSTMLayer_54219667145453
// MI455X (gfx1250) — compile-verified
//
#include <hip/hip_runtime.h>
#include <hip/hip_bf16.h>
#include <stdint.h>

// Problem constants (match reference)
#define B_   128
#define N_   8
#define M_   64
#define V_   512
#define K_   8
#define NP1  9           // N+1
#define NEG_INF_F (-1e30f)

#define THREADS 1024     // 32 wave32 per block: 4 waves per template row
#define WAVES   (THREADS / 32)

// ---------- CDNA5 async global<->LDS DMA (ASYNCcnt path) ----------
__device__ __forceinline__ void async_load_b128(unsigned lds_off, const void* gaddr) {
  // VDST = LDS byte offset (HW adds wave LDS_BASE), VADDR = 64-bit global addr
  asm volatile("global_load_async_to_lds_b128 %0, %1, off"
               :: "v"(lds_off), "v"((unsigned long long)(uintptr_t)gaddr)
               : "memory");
}

__device__ __forceinline__ void async_store_b128(const void* gaddr, unsigned lds_off) {
  asm volatile("global_store_async_from_lds_b128 %0, %1, off"
               :: "v"((unsigned long long)(uintptr_t)gaddr), "v"(lds_off)
               : "memory");
}

__device__ __forceinline__ void wait_async0() {
#if __has_builtin(__builtin_amdgcn_s_wait_asynccnt)
  __builtin_amdgcn_s_wait_asynccnt(0);
#else
  asm volatile("s_wait_asynccnt 0x0" ::: "memory");
#endif
}

// One workgroup per sample b. 1024 threads = 32 wave32.
// wave w handles template row k = w & 7, row-quarter q = w >> 3 (rows 16q..16q+15).
extern "C" __global__ void __launch_bounds__(THREADS)
scan_template_kernel(const float* __restrict__ decodings,      // [B,N,M,V]
                     const float* __restrict__ dec_sem_logits, // [7,21,K,N+1]
                     const float* __restrict__ gumbel,         // [B,K,N+1]
                     const int*   __restrict__ target_types,   // [B]
                     const int*   __restrict__ spans,          // [B]
                     float*       __restrict__ out)            // [B,M,V]
{
  extern __shared__ float4 smem4[];   // M_*V_/4 float4 = 128 KB output tile
  __shared__ int s_sel[K_], s_last[K_], s_idx[K_], s_olen[K_], s_total;

  const int b    = blockIdx.x;
  const int tid  = threadIdx.x;
  const int wave = tid >> 5;
  const int lane = tid & 31;
  const int k    = wave & 7;     // template row owned by this wave
  const int q    = wave >> 3;    // quarter of the M dimension

  const int ttype = target_types[b];
  const int span  = spans[b];

  if (tid < K_) s_last[tid] = -1;          // reduction identity for "last non-PAD row"

  // ---- Phase 1a: hard Gumbel sample for template row k ----
  // argmax(softmax((log_softmax(masked)+g)/tau)) == argmax(masked + g):
  // log_softmax subtracts a per-row constant; softmax/exp strictly monotone.
  // Every wave recomputes sel_k redundantly (18 tiny loads) to avoid a barrier.
  int sel = 0;
  if (lane == 0) {
    if (ttype == 20) {
      sel = (k == 0) ? 1 : 0;              // fixed start template for type 20
    } else {
      const float* lg = dec_sem_logits
          + ((size_t)(span - 2) * 21 + (size_t)(ttype - 9)) * (K_ * NP1)
          + (size_t)k * NP1;
      const float* gn = gumbel + (size_t)b * (K_ * NP1) + (size_t)k * NP1;
      float best = -3.4e38f; int bi = 0;
      for (int j = 0; j < NP1; ++j) {
        float lv = (j <= span) ? lg[j] : NEG_INF_F;
        float v  = lv + gn[j];
        if (v > best) { best = v; bi = j; } // strict > keeps first max (jnp.argmax)
      }
      sel = bi;
    }
    if (q == 0) s_sel[k] = sel;            // one writer per k
  }
  sel = __shfl(sel, 0);

  // ---- Phase 1b: this wave scans rows 16q..16q+15 of the selected decoding ----
  if (sel != 0) {
    const float4* src4 =
        (const float4*)(decodings + ((size_t)b * N_ + (size_t)(sel - 1)) * (M_ * V_));
    int localLast = -1;
    for (int mm = 0; mm < M_ / 4; ++mm) {
      const int m = q * (M_ / 4) + mm;
      const float4* row4 = src4 + m * (V_ / 4);
      // lane-local argmax over 16 contiguous floats (ascending index, strict >)
      float bv = -3.4e38f; int bidx = 0x7fffffff;
      #pragma unroll
      for (int c = 0; c < 4; ++c) {
        float4 qv = row4[lane * 4 + c];
        int base = (lane * 4 + c) * 4;
        if (qv.x > bv) { bv = qv.x; bidx = base + 0; }
        if (qv.y > bv) { bv = qv.y; bidx = base + 1; }
        if (qv.z > bv) { bv = qv.z; bidx = base + 2; }
        if (qv.w > bv) { bv = qv.w; bidx = base + 3; }
      }
      // wave32 butterfly: max value, lowest index on ties (matches jnp.argmax)
      #pragma unroll
      for (int off = 16; off >= 1; off >>= 1) {
        float ov = __shfl_xor(bv, off);
        int   oi = __shfl_xor(bidx, off);
        if (ov > bv || (ov == bv && oi < bidx)) { bv = ov; bidx = oi; }
      }
      if (bidx != 0) localLast = m;        // last position whose argmax != PAD(0)
    }
    if (lane == 0 && localLast >= 0) atomicMax(&s_last[k], localLast);  // ds_max_i32
  }
  __syncthreads();

  // ---- Phase 1c: saturated exclusive prefix -> segment placement ----
  if (tid == 0) {
    int excl = 0;
    for (int kk = 0; kk < K_; ++kk) {
      int ln  = s_last[kk] + 1;                       // lens (0 if no non-PAD row)
      int idx = (excl < M_) ? excl : M_;              // idx  = min(excl, M)
      int ol  = (ln < (M_ - idx)) ? ln : (M_ - idx);  // olen = min(len, M-idx)
      s_idx[kk] = idx; s_olen[kk] = ol;
      excl += ln;
    }
    s_total = (excl < M_) ? excl : M_;                // covered rows = [0, total)
  }
  __syncthreads();
  const int total = s_total;

  // ---- Phase 2a: assemble out[b] tile in LDS ----
  // zero the uncovered tail with DS stores
  float4 z; z.x = z.y = z.z = z.w = 0.0f;
  for (int i = total * (V_ / 4) + tid; i < M_ * (V_ / 4); i += THREADS) smem4[i] = z;

  // async-DMA covered rows global -> LDS (wave w handles rows w, w+32, ...)
  for (int r = wave; r < total; r += WAVES) {
    int kk = 0;
    #pragma unroll
    for (int t = 0; t < K_; ++t) {
      if (r >= s_idx[t] && r < s_idx[t] + s_olen[t]) kk = t;
    }
    const int src = s_sel[kk] - 1;       // segments only exist where sel != 0
    const int off = r - s_idx[kk];
    const float4* srcRow = (const float4*)(decodings
        + (((size_t)b * N_ + (size_t)src) * M_ + (size_t)off) * V_);
    unsigned ldsRow = (unsigned)(uintptr_t)(smem4 + r * (V_ / 4));
    #pragma unroll
    for (int c = 0; c < 4; ++c) {
      int f4 = c * 32 + lane;            // 128 float4 per 2 KB row
      async_load_b128(ldsRow + (unsigned)(f4 * 16), srcRow + f4);
    }
  }
  wait_async0();        // this wave's async loads have landed in LDS
  __syncthreads();      // + all DS zero-stores and other waves' DMAs visible

  // ---- Phase 2b: async-DMA the full 128 KB tile LDS -> out[b] ----
  float4* outTile = (float4*)(out + (size_t)b * (M_ * V_));
  for (int i = tid; i < M_ * (V_ / 4); i += THREADS) {
    async_store_b128(outTile + i, (unsigned)(uintptr_t)(smem4 + i));
  }
  wait_async0();        // drain ASYNCcnt before wave end
}

extern "C" void kernel_launch(void* const* d_in, const int* in_sizes, int n_in,
                              void* d_out, int out_size, void* d_ws, size_t ws_size,
                              hipStream_t stream) {
  const float* decodings      = (const float*)d_in[0];
  // d_in[1] (variables) is unused by the reference computation
  const float* dec_sem_logits = (const float*)d_in[2];
  const float* gumbel         = (const float*)d_in[3];
  const int*   target_types   = (const int*)d_in[4];
  const int*   spans          = (const int*)d_in[5];
  float*       out            = (float*)d_out;

  const size_t shmem = (size_t)M_ * V_ * sizeof(float);   // 128 KB tile (<= 320 KB/WGP)
  hipLaunchKernelGGL(scan_template_kernel, dim3(B_), dim3(THREADS), shmem, stream,
                     decodings, dec_sem_logits, gumbel, target_types, spans, out);
}